// GCNBlock_57286273794486
// MI455X (gfx1250) — compile-verified
//
#include <hip/hip_runtime.h>
#include <math.h>

#define DIM 128
#define RREL 2
#define LLAY 2

typedef __attribute__((ext_vector_type(2))) float v2f;
typedef __attribute__((ext_vector_type(8))) float v8f;

// ---------------------------------------------------------------------------
// Utility: zero-fill (graph-capture safe; no hipMemsetAsync needed)
// ---------------------------------------------------------------------------
__global__ void zero_f32(float* __restrict__ p, size_t n) {
  size_t i = (size_t)blockIdx.x * blockDim.x + threadIdx.x;
  size_t stride = (size_t)gridDim.x * blockDim.x;
  for (; i < n; i += stride) p[i] = 0.0f;
}

// ---------------------------------------------------------------------------
// In-degree per relation (layer-invariant): deg[r][dst] += 1
// ---------------------------------------------------------------------------
__global__ void degree_kernel(const int* __restrict__ edst,
                              float* __restrict__ deg,
                              int nNodes, int nEdges) {
  const int r = blockIdx.y;
  const int e = blockIdx.x * blockDim.x + threadIdx.x;
  if (e < nEdges) {
    const int t = edst[(size_t)r * nEdges + e];
    atomicAdd(&deg[(size_t)r * nNodes + t], 1.0f);
  }
}

// ---------------------------------------------------------------------------
// Edge scatter: agg[r][dst][:] += h[src][:]   (32 lanes x float4 per edge)
// ---------------------------------------------------------------------------
__global__ void scatter_kernel(const float* __restrict__ h,
                               const int* __restrict__ esrc,
                               const int* __restrict__ edst,
                               float* __restrict__ agg,
                               int nNodes, int nEdges) {
  const int r = blockIdx.y;
  const long long gid = (long long)blockIdx.x * blockDim.x + threadIdx.x;
  const int edge = (int)(gid >> 5);       // 32 threads per edge
  const int q = ((int)gid & 31) * 4;      // 4 floats per thread
  if (edge >= nEdges) return;
  const int s = esrc[(size_t)r * nEdges + edge];
  const int t = edst[(size_t)r * nEdges + edge];
  const float4 v = *(const float4*)(h + (size_t)s * DIM + q);
  float* dst = agg + ((size_t)r * nNodes + t) * DIM + q;
  atomicAdd(dst + 0, v.x);
  atomicAdd(dst + 1, v.y);
  atomicAdd(dst + 2, v.z);
  atomicAdd(dst + 3, v.w);
}

// ---------------------------------------------------------------------------
// Per-node: hr = (agg[r]/max(deg,1)) @ W[r] + b[r]; out = sum_r elu(hr)
// Block = 256 threads = 8 waves. Block owns 16 rows; each wave owns a 16-col
// tile. A slab (both relations, normalized) staged in LDS; WMMA f32 16x16x4.
// ---------------------------------------------------------------------------
__global__ __launch_bounds__(256) void gemm_elu_kernel(
    const float* __restrict__ agg,   // [R][N][D]
    const float* __restrict__ deg,   // [R][N]
    const float* __restrict__ W,     // [R][D][D] (this layer)
    const float* __restrict__ bias,  // [R][D]    (this layer)
    float* __restrict__ out,         // [N][D]
    int nNodes) {
  __shared__ float sA[RREL][16][DIM + 4];  // +4 pad: spread LDS banks

  const int tid = threadIdx.x;
  const int tileM = blockIdx.x * 16;

  // Stage normalized aggregates for both relations into LDS (coalesced).
  for (int r = 0; r < RREL; ++r) {
    for (int i = tid; i < 16 * DIM; i += 256) {
      const int row = i >> 7;          // DIM == 128
      const int d = i & (DIM - 1);
      const int node = tileM + row;
      float v = 0.0f;
      if (node < nNodes) {
        float dg = deg[(size_t)r * nNodes + node];
        dg = dg < 1.0f ? 1.0f : dg;
        v = agg[((size_t)r * nNodes + node) * DIM + d] / dg;
      }
      sA[r][row][d] = v;
    }
  }
  __syncthreads();

  const int wave = tid >> 5;
  const int lane = tid & 31;
  const int half = lane >> 4;      // K-half for A/B frags, M-half for C/D
  const int lane16 = lane & 15;    // A row / B,C,D column within tile
  const int tileN = wave * 16;

  const float* W0 = W;
  const float* W1 = W + (size_t)DIM * DIM;

  v8f c0 = {};
  v8f c1 = {};
  for (int k0 = 0; k0 < DIM; k0 += 4) {
    const int ka = k0 + half * 2;  // this lane's K base within the K=4 step
    v2f a0, a1, b0, b1;
    // A frag: A[M = lane16][K = ka + v]
    a0.x = sA[0][lane16][ka];
    a0.y = sA[0][lane16][ka + 1];
    a1.x = sA[1][lane16][ka];
    a1.y = sA[1][lane16][ka + 1];
    // B frag: B[K = ka + v][N = tileN + lane16]
    b0.x = W0[(size_t)ka * DIM + tileN + lane16];
    b0.y = W0[(size_t)(ka + 1) * DIM + tileN + lane16];
    b1.x = W1[(size_t)ka * DIM + tileN + lane16];
    b1.y = W1[(size_t)(ka + 1) * DIM + tileN + lane16];
    c0 = __builtin_amdgcn_wmma_f32_16x16x4_f32(false, a0, false, b0,
                                               (short)0, c0, false, false);
    c1 = __builtin_amdgcn_wmma_f32_16x16x4_f32(false, a1, false, b1,
                                               (short)0, c1, false, false);
  }

  // Epilogue: bias (per column), ELU, relation-sum, store.
  const int col = tileN + lane16;
  const float bs0 = bias[col];
  const float bs1 = bias[DIM + col];
#pragma unroll
  for (int v = 0; v < 8; ++v) {
    const int row = v + 8 * half;   // C/D layout: M = vgpr + 8*(lane/16)
    const int node = tileM + row;
    if (node < nNodes) {
      const float x0 = c0[v] + bs0;
      const float x1 = c1[v] + bs1;
      const float e0 = x0 > 0.0f ? x0 : (expf(x0) - 1.0f);
      const float e1 = x1 > 0.0f ? x1 : (expf(x1) - 1.0f);
      out[(size_t)node * DIM + col] = e0 + e1;
    }
  }
}

// ---------------------------------------------------------------------------
// Host orchestration
// ---------------------------------------------------------------------------
extern "C" void kernel_launch(void* const* d_in, const int* in_sizes, int n_in,
                              void* d_out, int out_size, void* d_ws, size_t ws_size,
                              hipStream_t stream) {
  const float* x = (const float*)d_in[0];     // [N][D]
  const float* W = (const float*)d_in[1];     // [L][R][D][D]
  const float* b = (const float*)d_in[2];     // [L][R][D]
  const int* esrc = (const int*)d_in[3];      // [R][E]
  const int* edst = (const int*)d_in[4];      // [R][E]
  float* out = (float*)d_out;                 // [N][D]

  const int nNodes = in_sizes[0] / DIM;
  const int nEdges = in_sizes[3] / RREL;

  // Workspace layout: h1 [N*D] | agg [R*N*D] | deg [R*N]   (~77 MB)
  float* h1 = (float*)d_ws;
  float* agg = h1 + (size_t)nNodes * DIM;
  float* deg = agg + (size_t)RREL * nNodes * DIM;

  // Degrees once (layer-invariant).
  zero_f32<<<1024, 256, 0, stream>>>(deg, (size_t)RREL * nNodes);
  {
    dim3 grid((nEdges + 255) / 256, RREL);
    degree_kernel<<<grid, 256, 0, stream>>>(edst, deg, nNodes, nEdges);
  }

  const float* hin = x;
  for (int l = 0; l < LLAY; ++l) {
    float* hout = (l == LLAY - 1) ? out : h1;

    zero_f32<<<8192, 256, 0, stream>>>(agg, (size_t)RREL * nNodes * DIM);

    {
      const long long work = (long long)nEdges * 32;  // 32 threads/edge
      dim3 grid((unsigned)((work + 255) / 256), RREL);
      scatter_kernel<<<grid, 256, 0, stream>>>(hin, esrc, edst, agg,
                                               nNodes, nEdges);
    }

    {
      dim3 grid((nNodes + 15) / 16);
      gemm_elu_kernel<<<grid, 256, 0, stream>>>(
          agg, deg,
          W + (size_t)l * RREL * DIM * DIM,
          b + (size_t)l * RREL * DIM,
          hout, nNodes);
    }

    hin = h1;
  }
}